// NetCostGNN_49606872269110
// MI455X (gfx1250) — compile-verified
//
#include <hip/hip_runtime.h>

typedef __attribute__((ext_vector_type(2))) float v2f;
typedef __attribute__((ext_vector_type(8))) float v8f;

#define DHID 128

__device__ __forceinline__ void atomic_add_f32(float* p, float v) {
    __hip_atomic_fetch_add(p, v, __ATOMIC_RELAXED, __HIP_MEMORY_SCOPE_AGENT);
}

// ---------------------------------------------------------------- zero fill
__global__ void k_zero(float* __restrict__ p, size_t n) {
    size_t i = (size_t)blockIdx.x * blockDim.x + threadIdx.x;
    size_t stride = (size_t)gridDim.x * blockDim.x;
    for (; i < n; i += stride) p[i] = 0.0f;
}

// ---------------------------------------------------------------- degree
__global__ void k_count_deg(const int* __restrict__ dst, float* __restrict__ deg, int E) {
    int i = blockIdx.x * blockDim.x + threadIdx.x;
    if (i < E) atomic_add_f32(&deg[dst[i]], 1.0f);
}

__global__ void k_inv_deg(float* __restrict__ deg, int n) {
    int i = blockIdx.x * blockDim.x + threadIdx.x;
    if (i < n) deg[i] = 1.0f / fmaxf(deg[i], 1.0f);
}

// ---------------------------------------------------------------- scatter-add
// One wave per edge; 32 lanes x float4 = 128 features. Hardware f32 atomics
// land in L2 (the whole 25.6MB feature matrix is resident in MI455X's 192MB L2).
__global__ void k_scatter(const float* __restrict__ feat, const int* __restrict__ src,
                          const int* __restrict__ dst, float* __restrict__ agg, int E) {
    int wave = (blockIdx.x * blockDim.x + threadIdx.x) >> 5;
    int lane = threadIdx.x & 31;
    if (wave >= E) return;
    int s = src[wave];
    int d = dst[wave];
    const float4 v = *(const float4*)(feat + (size_t)s * DHID + lane * 4);
    float* p = agg + (size_t)d * DHID + lane * 4;
    atomic_add_f32(p + 0, v.x);
    atomic_add_f32(p + 1, v.y);
    atomic_add_f32(p + 2, v.z);
    atomic_add_f32(p + 3, v.w);
}

// ---------------------------------------------------------------- SAGE GEMM (fp32 WMMA)
// h[m,n] = relu( sum_k (agg[m,k]*inv_deg[m]) * Wl[k,n] + xin[m,k] * Wr[k,n] + bias[n] )
// One wave computes a 16x32 output strip (two 16x16 tiles) so the A operands
// (agg, xin rows) are loaded once and fed to 4 V_WMMA_F32_16X16X4_F32 per k-step.
//   A layout (16x4 f32): lane l: row = l&15, koff = (l>>4)*2; v0=A[row][k+koff], v1=A[row][k+koff+1]
//   B layout (4x16 f32): lane l: col = l&15, koff = (l>>4)*2; v0=B[k+koff][col], v1=B[k+koff+1][col]
//   C/D (16x16 f32):     lane l: col = l&15, row = vgpr + (l>>4)*8
__global__ void k_sage_gemm(const float* __restrict__ agg, const float* __restrict__ invdeg,
                            const float* __restrict__ xin, const float* __restrict__ Wl,
                            const float* __restrict__ bias, const float* __restrict__ Wr,
                            float* __restrict__ hout, int nrows) {
    const int wave = (blockIdx.x * blockDim.x + threadIdx.x) >> 5;
    const int lane = threadIdx.x & 31;
    const int npair = DHID / 32;                   // 4 column-pairs of 16x16 tiles
    const int tile_m = wave / npair;
    const int pair   = wave % npair;
    if (tile_m * 16 >= nrows) return;              // whole wave in/out -> EXEC stays full

    const int half = lane >> 4;                    // 0 or 1
    const int lid  = lane & 15;
    const int koff = half * 2;
    int row = tile_m * 16 + lid;
    if (row >= nrows) row = nrows - 1;             // clamp loads, keep EXEC all-1s
    const int col0 = pair * 32 + lid;
    const int col1 = col0 + 16;

    const float idg = invdeg[row];
    const float* __restrict__ ap  = agg + (size_t)row * DHID + koff;
    const float* __restrict__ xp  = xin + (size_t)row * DHID + koff;
    const float* __restrict__ bl0p = Wl + (size_t)koff * DHID + col0;
    const float* __restrict__ bl1p = Wl + (size_t)koff * DHID + col1;
    const float* __restrict__ br0p = Wr + (size_t)koff * DHID + col0;
    const float* __restrict__ br1p = Wr + (size_t)koff * DHID + col1;

    v8f acc0 = {};
    v8f acc1 = {};
#pragma unroll 2
    for (int k = 0; k < DHID; k += 4) {
        const size_t r0 = (size_t)k * DHID;
        const size_t r1 = (size_t)(k + 1) * DHID;
        v2f am, ax, bl0, bl1, br0, br1;
        am.x = ap[k] * idg;
        am.y = ap[k + 1] * idg;
        ax.x = xp[k];
        ax.y = xp[k + 1];
        bl0.x = bl0p[r0];  bl0.y = bl0p[r1];
        bl1.x = bl1p[r0];  bl1.y = bl1p[r1];
        br0.x = br0p[r0];  br0.y = br0p[r1];
        br1.x = br1p[r0];  br1.y = br1p[r1];
        acc0 = __builtin_amdgcn_wmma_f32_16x16x4_f32(false, am, false, bl0,
                                                     (short)0, acc0, false, false);
        acc1 = __builtin_amdgcn_wmma_f32_16x16x4_f32(false, am, false, bl1,
                                                     (short)0, acc1, false, false);
        acc0 = __builtin_amdgcn_wmma_f32_16x16x4_f32(false, ax, false, br0,
                                                     (short)0, acc0, false, false);
        acc1 = __builtin_amdgcn_wmma_f32_16x16x4_f32(false, ax, false, br1,
                                                     (short)0, acc1, false, false);
    }

    const float bn0 = bias[col0];
    const float bn1 = bias[col1];
    const int mbase = tile_m * 16 + half * 8;
    float* __restrict__ o0 = hout + (size_t)mbase * DHID + col0;
    float* __restrict__ o1 = hout + (size_t)mbase * DHID + col1;

    if (tile_m * 16 + 16 <= nrows) {
        // full tile: branch-free stores (the only path when N % 16 == 0)
#pragma unroll
        for (int v = 0; v < 8; ++v) {
            float r0v = acc0[v] + bn0;
            float r1v = acc1[v] + bn1;
            o0[(size_t)v * DHID] = r0v > 0.0f ? r0v : 0.0f;
            o1[(size_t)v * DHID] = r1v > 0.0f ? r1v : 0.0f;
        }
    } else {
#pragma unroll
        for (int v = 0; v < 8; ++v) {
            if (mbase + v < nrows) {
                float r0v = acc0[v] + bn0;
                float r1v = acc1[v] + bn1;
                o0[(size_t)v * DHID] = r0v > 0.0f ? r0v : 0.0f;
                o1[(size_t)v * DHID] = r1v > 0.0f ? r1v : 0.0f;
            }
        }
    }
}

// ---------------------------------------------------------------- output projection
// out[i] = dot(h[i,:], Wout[:,0]) + b_out; one wave per node, shfl reduction.
__global__ void k_out_proj(const float* __restrict__ h, const float* __restrict__ Wout,
                           const float* __restrict__ bout, float* __restrict__ out, int n) {
    int wave = (blockIdx.x * blockDim.x + threadIdx.x) >> 5;
    int lane = threadIdx.x & 31;
    if (wave >= n) return;
    const float4 hv = *(const float4*)(h + (size_t)wave * DHID + lane * 4);
    const float4 wv = *(const float4*)(Wout + lane * 4);
    float p = hv.x * wv.x + hv.y * wv.y + hv.z * wv.z + hv.w * wv.w;
#pragma unroll
    for (int off = 16; off > 0; off >>= 1) p += __shfl_xor(p, off, 32);
    if (lane == 0) out[wave] = p + bout[0];
}

// ---------------------------------------------------------------- launcher
extern "C" void kernel_launch(void* const* d_in, const int* in_sizes, int n_in,
                              void* d_out, int out_size, void* d_ws, size_t ws_size,
                              hipStream_t stream) {
    const float* x    = (const float*)d_in[0];
    const int*   ei   = (const int*)d_in[1];
    const float* Wl1  = (const float*)d_in[2];
    const float* bl1  = (const float*)d_in[3];
    const float* Wr1  = (const float*)d_in[4];
    const float* Wl2  = (const float*)d_in[5];
    const float* bl2  = (const float*)d_in[6];
    const float* Wr2  = (const float*)d_in[7];
    const float* Wout = (const float*)d_in[8];
    const float* bout = (const float*)d_in[9];

    const int N = in_sizes[0] / DHID;
    const int E = in_sizes[1] / 2;
    const int* src = ei;          // edge_index[0]
    const int* dst = ei + E;      // edge_index[1]

    float* out  = (float*)d_out;         // [N]
    float* hout = (float*)d_out + N;     // [N,128] (second tuple element)

    float* ws   = (float*)d_ws;
    size_t npad = ((size_t)N + 255) & ~(size_t)255;
    float* deg  = ws;                    // [N] -> becomes inv_deg in place
    float* agg  = ws + npad;             // [N*128]
    float* h1   = agg + (size_t)N * DHID;// [N*128]

    const dim3 blk(256);
    const dim3 gZeroN(512);
    const dim3 gZeroAgg(4096);
    const dim3 gDeg((E + 255) / 256);
    const dim3 gInv((N + 255) / 256);
    const dim3 gScat((int)(((size_t)E * 32 + 255) / 256));
    const int  tiles_m = (N + 15) / 16;
    const dim3 gGemm((int)(((size_t)tiles_m * (DHID / 32) * 32 + 255) / 256));
    const dim3 gOut((int)(((size_t)N * 32 + 255) / 256));

    // degrees (shared by both layers)
    k_zero<<<gZeroN, blk, 0, stream>>>(deg, (size_t)N);
    k_count_deg<<<gDeg, blk, 0, stream>>>(dst, deg, E);
    k_inv_deg<<<gInv, blk, 0, stream>>>(deg, N);

    // layer 1
    k_zero<<<gZeroAgg, blk, 0, stream>>>(agg, (size_t)N * DHID);
    k_scatter<<<gScat, blk, 0, stream>>>(x, src, dst, agg, E);
    k_sage_gemm<<<gGemm, blk, 0, stream>>>(agg, deg, x, Wl1, bl1, Wr1, h1, N);

    // layer 2
    k_zero<<<gZeroAgg, blk, 0, stream>>>(agg, (size_t)N * DHID);
    k_scatter<<<gScat, blk, 0, stream>>>(h1, src, dst, agg, E);
    k_sage_gemm<<<gGemm, blk, 0, stream>>>(agg, deg, h1, Wl2, bl2, Wr2, hout, N);

    // output head
    k_out_proj<<<gOut, blk, 0, stream>>>(hout, Wout, bout, out, N);
}